// GraphAttentionLayer_47132971106801
// MI455X (gfx1250) — compile-verified
//
#include <hip/hip_runtime.h>
#include <math.h>
#include <stdint.h>

typedef float v2f __attribute__((ext_vector_type(2)));
typedef float v8f __attribute__((ext_vector_type(8)));

#define N_NODES 4096
#define IN_F    512
#define OUT_F   256
#define ALPHA   0.2f
#define NEG_INF -9.0e15f
#define KC      256          // K-chunk staged in LDS for the big GEMM

// ---------------------------------------------------------------------------
// Kernel 1: Wh = h @ W   [4096,512] @ [512,256] -> [4096,256], fp32 WMMA
// Block = 512 threads = 16 waves; wave w -> 16x16 tile (rows 16*bx, cols 16*w).
// ---------------------------------------------------------------------------
__global__ __launch_bounds__(512) void gemm_wh_kernel(
    const float* __restrict__ h, const float* __restrict__ w,
    float* __restrict__ wh) {
  const int lane = threadIdx.x & 31;
  const int wave = threadIdx.x >> 5;
  const int row0 = blockIdx.x * 16;
  const int col0 = wave * 16;
  const int m    = lane & 15;
  const int koff = (lane >> 4) << 1;

  v8f c = {};
  const float* arow = h + (size_t)(row0 + m) * IN_F + koff;
  const float* bcol = w + (size_t)koff * OUT_F + col0 + m;

#pragma unroll 8
  for (int k = 0; k < IN_F; k += 4) {
    v2f a, b;
    a.x = arow[k];
    a.y = arow[k + 1];
    b.x = bcol[(size_t)k * OUT_F];
    b.y = bcol[(size_t)(k + 1) * OUT_F];
    c = __builtin_amdgcn_wmma_f32_16x16x4_f32(false, a, false, b,
                                              (short)0, c, false, false);
  }

  const int rbase = (lane >> 4) << 3;
#pragma unroll
  for (int r = 0; r < 8; ++r) {
    wh[(size_t)(row0 + rbase + r) * OUT_F + col0 + m] = c[r];
  }
}

// ---------------------------------------------------------------------------
// Kernel 2: f1[i] = Wh[i,:] . a[:256],  f2[i] = Wh[i,:] . a[256:]
// ---------------------------------------------------------------------------
__global__ __launch_bounds__(256) void fvec_kernel(
    const float* __restrict__ wh, const float* __restrict__ a,
    float* __restrict__ f1, float* __restrict__ f2) {
  const int i = blockIdx.x * blockDim.x + threadIdx.x;
  if (i >= N_NODES) return;
  const float* r = wh + (size_t)i * OUT_F;
  float s1 = 0.f, s2 = 0.f;
#pragma unroll 4
  for (int j = 0; j < OUT_F; j += 4) {
    float4 v  = *(const float4*)(r + j);
    float4 a1 = *(const float4*)(a + j);
    float4 a2 = *(const float4*)(a + OUT_F + j);
    s1 += v.x * a1.x + v.y * a1.y + v.z * a1.z + v.w * a1.w;
    s2 += v.x * a2.x + v.y * a2.y + v.z * a2.z + v.w * a2.w;
  }
  f1[i] = s1;
  f2[i] = s2;
}

// ---------------------------------------------------------------------------
// Kernel 3: fused mask + row softmax; one block (256 threads) per row;
// 16 KB row cached in LDS so adj is read exactly once from HBM.
// ---------------------------------------------------------------------------
__global__ __launch_bounds__(256) void softmax_row_kernel(
    const int* __restrict__ adj, const float* __restrict__ f1,
    const float* __restrict__ f2, float* __restrict__ attn) {
  __shared__ float e_row[N_NODES];
  __shared__ float red[256];

  const int i = blockIdx.x;
  const int t = threadIdx.x;
  const float f1i = f1[i];
  const int* arow = adj + (size_t)i * N_NODES;

  float lmax = NEG_INF;
  for (int j = t; j < N_NODES; j += 256) {
    float e;
    if (arow[j] > 0) {
      float x = f1i + f2[j];
      e = (x > 0.f) ? x : ALPHA * x;
    } else {
      e = NEG_INF;
    }
    e_row[j] = e;
    lmax = fmaxf(lmax, e);
  }
  red[t] = lmax;
  __syncthreads();
  for (int s = 128; s > 0; s >>= 1) {
    if (t < s) red[t] = fmaxf(red[t], red[t + s]);
    __syncthreads();
  }
  const float rmax = red[0];
  __syncthreads();

  float lsum = 0.f;
  for (int j = t; j < N_NODES; j += 256) {
    float ex = __expf(e_row[j] - rmax);   // all-masked row -> exp(0)=1, matches ref
    e_row[j] = ex;
    lsum += ex;
  }
  red[t] = lsum;
  __syncthreads();
  for (int s = 128; s > 0; s >>= 1) {
    if (t < s) red[t] += red[t + s];
    __syncthreads();
  }
  const float inv = 1.0f / red[0];
  __syncthreads();

  float* orow = attn + (size_t)i * N_NODES;
  for (int j = t; j < N_NODES; j += 256) {
    orow[j] = e_row[j] * inv;
  }
}

// ---------------------------------------------------------------------------
// Kernel 4: h_prime = attention @ Wh (+ elu).  [4096,4096] @ [4096,256].
// A-panel (16 rows x KC) is staged into LDS with GLOBAL_LOAD_ASYNC_TO_LDS_B128
// (ASYNCcnt), double-buffered, so all 16 waves share one global read of the
// strip and the WMMA feed comes from LDS (ds_load_b64; unpadded rows, so the 16 rows hit the same banks).
// ---------------------------------------------------------------------------
__global__ __launch_bounds__(512) void gemm_attn_kernel(
    const float* __restrict__ attn, const float* __restrict__ wh,
    float* __restrict__ out) {
  __shared__ float ablk[2][16][KC];   // 2 x 16 KB double buffer

  const int t    = threadIdx.x;
  const int lane = t & 31;
  const int wave = t >> 5;
  const int row0 = blockIdx.x * 16;
  const int col0 = wave * 16;
  const int m    = lane & 15;
  const int koff = (lane >> 4) << 1;

  const float* bcol = wh + (size_t)koff * OUT_F + col0 + m;

  // --- async stage of one 16 x KC chunk: 1024 float4, 2 per thread ---
  auto stage = [&](int buf, int kc) {
#pragma unroll
    for (int it = 0; it < 2; ++it) {
      const int f  = t + it * 512;        // float4 index in [0,1024)
      const int r  = f >> 6;              // row 0..15
      const int kq = f & 63;              // float4 within row
      const float* g = attn + (size_t)(row0 + r) * N_NODES + kc + kq * 4;
      uint32_t lds_off = (uint32_t)(uintptr_t)&ablk[buf][r][kq * 4];
      uint64_t gaddr   = (uint64_t)(uintptr_t)g;
      asm volatile("global_load_async_to_lds_b128 %0, %1, off"
                   :: "v"(lds_off), "v"(gaddr) : "memory");
    }
  };

  v8f c = {};
  stage(0, 0);
  asm volatile("s_wait_asynccnt 0" ::: "memory");
  __syncthreads();

  int buf = 0;
  for (int ch = 0; ch < N_NODES / KC; ++ch) {
    if (ch + 1 < N_NODES / KC) stage(buf ^ 1, (ch + 1) * KC);

    const float* ar = &ablk[buf][m][koff];
    const int kc = ch * KC;
#pragma unroll 4
    for (int kk = 0; kk < KC; kk += 4) {
      v2f a, b;
      a.x = ar[kk];
      a.y = ar[kk + 1];
      b.x = bcol[(size_t)(kc + kk) * OUT_F];
      b.y = bcol[(size_t)(kc + kk + 1) * OUT_F];
      c = __builtin_amdgcn_wmma_f32_16x16x4_f32(false, a, false, b,
                                                (short)0, c, false, false);
    }

    asm volatile("s_wait_asynccnt 0" ::: "memory");
    __syncthreads();
    buf ^= 1;
  }

  const int rbase = (lane >> 4) << 3;
#pragma unroll
  for (int r = 0; r < 8; ++r) {
    float x = c[r];
    float y = (x > 0.f) ? x : expm1f(x);   // elu
    out[(size_t)(row0 + rbase + r) * OUT_F + col0 + m] = y;
  }
}

// ---------------------------------------------------------------------------
// Launch: inputs = {h[4096*512], adj[4096*4096] i32, weight[512*256], a[512]}
// d_out = elu(h_prime) [4096*256] ++ attention [4096*4096]
// d_ws  = Wh [4096*256] ++ f1 [4096] ++ f2 [4096]
// ---------------------------------------------------------------------------
extern "C" void kernel_launch(void* const* d_in, const int* in_sizes, int n_in,
                              void* d_out, int out_size, void* d_ws, size_t ws_size,
                              hipStream_t stream) {
  const float* h   = (const float*)d_in[0];
  const int*   adj = (const int*)d_in[1];
  const float* w   = (const float*)d_in[2];
  const float* a   = (const float*)d_in[3];

  float* out_elu = (float*)d_out;
  float* attn    = out_elu + (size_t)N_NODES * OUT_F;

  float* wh = (float*)d_ws;
  float* f1 = wh + (size_t)N_NODES * OUT_F;
  float* f2 = f1 + N_NODES;

  gemm_wh_kernel<<<N_NODES / 16, 512, 0, stream>>>(h, w, wh);
  fvec_kernel<<<N_NODES / 256, 256, 0, stream>>>(wh, a, f1, f2);
  softmax_row_kernel<<<N_NODES, 256, 0, stream>>>(adj, f1, f2, attn);
  gemm_attn_kernel<<<N_NODES / 16, 512, 0, stream>>>(attn, wh, out_elu);
}